// GMReader2ConvSortPoolReadout_86303072845941
// MI455X (gfx1250) — compile-verified
//
#include <hip/hip_runtime.h>
#include <cstdint>

// ---------------- problem constants ----------------
constexpr int G_   = 256;
constexpr int NPG_ = 512;
constexpr int NN_  = G_ * NPG_;          // 131072 nodes
constexpr int E_   = NN_ * 8;            // 1048576 edges
constexpr int IN_  = 64;
constexpr int H_   = 128;
constexpr int H2_  = 32;
constexpr int K_   = 64;
constexpr int OUT_ = 32;
constexpr int CIN_ = K_ * H_ + K_ * H2_; // 10240
#define NEG_SLOPE 0.01f
#define EPS_GN 1e-5f

typedef float v2f __attribute__((ext_vector_type(2)));
typedef float v8f __attribute__((ext_vector_type(8)));

// ---------------- utility kernels ----------------
__global__ void fill0_kernel(float* p, long long n) {
  long long i = (long long)blockIdx.x * blockDim.x + threadIdx.x;
  long long stride = (long long)gridDim.x * blockDim.x;
  for (; i < n; i += stride) p[i] = 0.0f;
}

__global__ void degree_kernel(const int* __restrict__ src, const int* __restrict__ dst,
                              float* __restrict__ dsrc, float* __restrict__ ddst) {
  int e = blockIdx.x * blockDim.x + threadIdx.x;
  if (e >= E_) return;
  atomicAdd(&dsrc[src[e]], 1.0f);
  atomicAdd(&ddst[dst[e]], 1.0f);
}

__global__ void rsqrt_kernel(float* __restrict__ v, int n) {
  int i = blockIdx.x * blockDim.x + threadIdx.x;
  if (i < n) v[i] = rsqrtf(fmaxf(v[i], 1.0f));
}

template<int C>
__global__ void scale_rows_kernel(const float* __restrict__ x, const float* __restrict__ s,
                                  float* __restrict__ y) {
  int i = blockIdx.x * blockDim.x + threadIdx.x;
  if (i >= NN_ * C) return;
  y[i] = x[i] * s[i / C];
}

// message scatter: agg[dst] += msg[src] * ew   (C channels per edge)
template<int C>
__global__ void scatter_kernel(const float* __restrict__ msg, const int* __restrict__ src,
                               const int* __restrict__ dst, const float* __restrict__ ew,
                               float* __restrict__ agg) {
  long long i = (long long)blockIdx.x * blockDim.x + threadIdx.x;
  int e = (int)(i / C);
  int c = (int)(i % C);
  if (e >= E_) return;
  float v = msg[(size_t)src[e] * C + c] * ew[e];
  atomicAdd(&agg[(size_t)dst[e] * C + c], v);
}

// ---------------- WMMA f32 GEMM: C[M x N] = diag(rowScale) * A[M x K] @ B[K x N] ----------------
// One wave computes a 16-row strip across all N columns (A fragment reused N/16 times).
template<int N, int K>
__global__ __launch_bounds__(256) void gemm_wmma_kernel(
    const float* __restrict__ A, const float* __restrict__ B,
    float* __restrict__ C, const float* __restrict__ rowScale, int M) {
  constexpr int NT = N / 16;
  int wave = blockIdx.x * (blockDim.x >> 5) + (threadIdx.x >> 5);
  int rowBase = wave << 4;
  if (rowBase >= M) return;
  int lane = threadIdx.x & 31;
  int mr = lane & 15;              // row (A) / col (B,C) within tile
  int kh = (lane >> 4) << 1;       // K pair base for this half-wave
  float s = rowScale ? rowScale[rowBase + mr] : 1.0f;
  const float* aRow = A + (size_t)(rowBase + mr) * K;
  v8f acc[NT] = {};
  for (int kb = 0; kb < K; kb += 4) {
    v2f a;
    a.x = aRow[kb + kh] * s;
    a.y = aRow[kb + kh + 1] * s;
#pragma unroll
    for (int t = 0; t < NT; ++t) {
      v2f b;
      b.x = B[(size_t)(kb + kh) * N + t * 16 + mr];
      b.y = B[(size_t)(kb + kh + 1) * N + t * 16 + mr];
      acc[t] = __builtin_amdgcn_wmma_f32_16x16x4_f32(
          false, a, false, b, (short)0, acc[t], false, false);
    }
  }
  int rOff = (lane >> 4) << 3;
#pragma unroll
  for (int t = 0; t < NT; ++t)
#pragma unroll
    for (int r = 0; r < 8; ++r)
      C[(size_t)(rowBase + r + rOff) * N + t * 16 + mr] = acc[t][r];
}

// ---------------- GraphNorm + leaky ----------------
// One block per graph; 256 threads; F channels, 256/F threads per channel.
template<int F>
__global__ __launch_bounds__(256) void graph_norm_leaky_kernel(
    const float* __restrict__ x, const float* __restrict__ rowScale,
    const float* __restrict__ gamma, const float* __restrict__ beta,
    const float* __restrict__ alpha, float* __restrict__ out) {
  constexpr int TPC = 256 / F;       // threads per channel
  constexpr int NPT = NPG_ / TPC;    // nodes per thread
  __shared__ float red[256];
  __shared__ float meanS[F];
  __shared__ float invS[F];
  int g = blockIdx.x;
  int t = threadIdx.x;
  int c = t % F;
  int seg = t / F;
  const float* xg = x + (size_t)g * NPG_ * F;
  const float* rs = rowScale ? rowScale + (size_t)g * NPG_ : nullptr;

  float sum = 0.0f;
  for (int n = seg * NPT; n < (seg + 1) * NPT; ++n) {
    float v = xg[(size_t)n * F + c];
    if (rs) v *= rs[n];
    sum += v;
  }
  red[t] = sum; __syncthreads();
  if (seg == 0) {
    float m = 0.0f;
    for (int k = 0; k < TPC; ++k) m += red[c + k * F];
    meanS[c] = m * (1.0f / NPG_);
  }
  __syncthreads();
  float am = alpha[c] * meanS[c];

  float vs = 0.0f;
  for (int n = seg * NPT; n < (seg + 1) * NPT; ++n) {
    float v = xg[(size_t)n * F + c];
    if (rs) v *= rs[n];
    float d = v - am;
    vs += d * d;
  }
  red[t] = vs; __syncthreads();
  if (seg == 0) {
    float m = 0.0f;
    for (int k = 0; k < TPC; ++k) m += red[c + k * F];
    invS[c] = rsqrtf(m * (1.0f / NPG_) + EPS_GN);
  }
  __syncthreads();

  float gm = gamma[c], bt = beta[c], inv = invS[c];
  float* og = out + (size_t)g * NPG_ * F;
  for (int n = seg * NPT; n < (seg + 1) * NPT; ++n) {
    float v = xg[(size_t)n * F + c];
    if (rs) v *= rs[n];
    float o = gm * (v - am) * inv + bt;
    og[(size_t)n * F + c] = (o >= 0.0f) ? o : NEG_SLOPE * o;
  }
}

// ---------------- per-node ascending bitonic channel sort (F power of 2) ----------------
template<int F>
__global__ void sort_channels_kernel(const float* __restrict__ x,
                                     float* __restrict__ xs, float* __restrict__ key) {
  __shared__ float s[F];
  int node = blockIdx.x;
  int t = threadIdx.x;
  s[t] = x[(size_t)node * F + t];
  __syncthreads();
  for (int size = 2; size <= F; size <<= 1) {
    for (int stride = size >> 1; stride > 0; stride >>= 1) {
      int j = t ^ stride;
      if (j > t) {
        bool dir = ((t & size) == 0);  // true => ascending segment
        float a = s[t], b = s[j];
        if ((b < a) == dir) { s[t] = b; s[j] = a; }
      }
      __syncthreads();
    }
  }
  xs[(size_t)node * F + t] = s[t];
  if (t == F - 1) key[node] = s[t];  // per-node max (last of ascending sort)
}

// ---------------- per-graph top-K by key (desc, tie -> lower index) ----------------
__global__ __launch_bounds__(512) void topk_kernel(const float* __restrict__ key,
                                                   int* __restrict__ topidx) {
  __shared__ float sv[NPG_];
  __shared__ int   si[NPG_];
  int g = blockIdx.x, t = threadIdx.x;
  sv[t] = key[g * NPG_ + t];
  si[t] = t;
  __syncthreads();
  for (int size = 2; size <= NPG_; size <<= 1) {
    for (int stride = size >> 1; stride > 0; stride >>= 1) {
      int j = t ^ stride;
      if (j > t) {
        bool dir = ((t & size) == 0);  // true => descending segment
        float av = sv[t], bv = sv[j];
        int ai = si[t], bi = si[j];
        bool jFirst = (bv > av) || (bv == av && bi < ai);
        if (jFirst == dir) { sv[t] = bv; sv[j] = av; si[t] = bi; si[j] = ai; }
      }
      __syncthreads();
    }
  }
  if (t < K_) topidx[g * K_ + t] = si[t];
}

template<int F>
__global__ void gather_topk_kernel(const float* __restrict__ hs, const int* __restrict__ topidx,
                                   float* __restrict__ readout, int colOff) {
  int i = blockIdx.x * blockDim.x + threadIdx.x;
  if (i >= G_ * K_ * F) return;
  int c = i % F;
  int k = (i / F) % K_;
  int g = i / (F * K_);
  int node = g * NPG_ + topidx[g * K_ + k];
  readout[(size_t)g * CIN_ + colOff + k * F + c] = hs[(size_t)node * F + c];
}

// ---------------- classifier: out[256x32] += R[256x10240] @ Wc^T (split-K, atomics) ----------------
__global__ __launch_bounds__(256) void clf_wmma_kernel(const float* __restrict__ R,
                                                       const float* __restrict__ Wc,
                                                       float* __restrict__ out) {
  constexpr int Kc = CIN_, Nc = OUT_;
  constexpr int CH = 256;            // K per chunk
  constexpr int NCH = Kc / CH;       // 40 chunks
  int wave = blockIdx.x * (blockDim.x >> 5) + (threadIdx.x >> 5);
  int rt = wave / NCH;               // 0..15 row tiles (M=256)
  int ck = wave % NCH;
  if (rt >= G_ / 16) return;
  int rowBase = rt << 4;
  int lane = threadIdx.x & 31;
  int mr = lane & 15;
  int kh = (lane >> 4) << 1;
  const float* aRow = R + (size_t)(rowBase + mr) * Kc;
  v8f acc[2] = {};
  for (int kb = ck * CH; kb < ck * CH + CH; kb += 4) {
    v2f a;
    a.x = aRow[kb + kh];
    a.y = aRow[kb + kh + 1];
#pragma unroll
    for (int t = 0; t < 2; ++t) {
      v2f b;  // B[k][n] = Wc[n*K + k]
      b.x = Wc[(size_t)(t * 16 + mr) * Kc + kb + kh];
      b.y = Wc[(size_t)(t * 16 + mr) * Kc + kb + kh + 1];
      acc[t] = __builtin_amdgcn_wmma_f32_16x16x4_f32(
          false, a, false, b, (short)0, acc[t], false, false);
    }
  }
  int rOff = (lane >> 4) << 3;
#pragma unroll
  for (int t = 0; t < 2; ++t)
#pragma unroll
    for (int r = 0; r < 8; ++r)
      atomicAdd(&out[(size_t)(rowBase + r + rOff) * Nc + t * 16 + mr], acc[t][r]);
}

// ---------------- host launcher ----------------
extern "C" void kernel_launch(void* const* d_in, const int* in_sizes, int n_in,
                              void* d_out, int out_size, void* d_ws, size_t ws_size,
                              hipStream_t stream) {
  const float* features = (const float*)d_in[0];
  const float* ew       = (const float*)d_in[1];
  const float* W1       = (const float*)d_in[2];
  const float* W2       = (const float*)d_in[3];
  const float* gamma1   = (const float*)d_in[4];
  const float* beta1    = (const float*)d_in[5];
  const float* alpha1   = (const float*)d_in[6];
  const float* gamma2   = (const float*)d_in[7];
  const float* beta2    = (const float*)d_in[8];
  const float* alpha2   = (const float*)d_in[9];
  const float* Wc       = (const float*)d_in[10];
  const int*   esrc     = (const int*)d_in[11];
  const int*   edst     = (const int*)d_in[12];
  float* out = (float*)d_out;

  // workspace layout (floats)
  float* ws      = (float*)d_ws;
  float* inv_src = ws;                       // NN
  float* inv_dst = inv_src + NN_;            // NN
  float* R0      = inv_dst + NN_;            // NN*64: fs -> (t2 | agg2)
  float* R1      = R0 + (size_t)NN_ * 64;    // NN*64: agg1 -> (h2 | hsorted2)
  float* R2      = R1 + (size_t)NN_ * 64;    // NN*128: x1g -> hsorted1
  float* R3      = R2 + (size_t)NN_ * 128;   // NN*128: h1
  float* key1    = R3 + (size_t)NN_ * 128;   // NN
  float* key2    = key1 + NN_;               // NN
  int*   topidx1 = (int*)(key2 + NN_);       // G*K
  int*   topidx2 = topidx1 + G_ * K_;        // G*K
  float* readout = (float*)(topidx2 + G_ * K_); // G*CIN

  float* t2   = R0;                 // NN*32
  float* agg2 = R0 + (size_t)NN_ * 32;
  float* h2   = R1;                 // NN*32
  float* hs2  = R1 + (size_t)NN_ * 32;

  // --- degree norms ---
  fill0_kernel<<<2048, 256, 0, stream>>>(inv_src, 2LL * NN_);
  fill0_kernel<<<2048, 256, 0, stream>>>(R1, (long long)NN_ * 64);  // agg1 = 0
  degree_kernel<<<E_ / 256, 256, 0, stream>>>(esrc, edst, inv_src, inv_dst);
  rsqrt_kernel<<<(2 * NN_) / 256, 256, 0, stream>>>(inv_src, 2 * NN_);

  // --- Conv1: scale -> scatter -> (inv_dst row-scaled) GEMM ---
  scale_rows_kernel<IN_><<<(NN_ * IN_) / 256, 256, 0, stream>>>(features, inv_src, R0);
  scatter_kernel<IN_><<<(int)(((long long)E_ * IN_) / 256), 256, 0, stream>>>(R0, esrc, edst, ew, R1);
  gemm_wmma_kernel<H_, IN_><<<NN_ / 128, 256, 0, stream>>>(R1, W1, R2, inv_dst, NN_);

  // --- Norm1 + leaky -> h1 ---
  graph_norm_leaky_kernel<H_><<<G_, 256, 0, stream>>>(R2, nullptr, gamma1, beta1, alpha1, R3);

  // --- SortPool1 ---
  sort_channels_kernel<H_><<<NN_, H_, 0, stream>>>(R3, R2, key1);
  topk_kernel<<<G_, NPG_, 0, stream>>>(key1, topidx1);
  gather_topk_kernel<H_><<<(G_ * K_ * H_) / 256, 256, 0, stream>>>(R2, topidx1, readout, 0);

  // --- Conv2: (inv_src row-scaled) GEMM -> scatter -> inv_dst folded into norm2 ---
  gemm_wmma_kernel<H2_, H_><<<NN_ / 128, 256, 0, stream>>>(R3, W2, t2, inv_src, NN_);
  fill0_kernel<<<2048, 256, 0, stream>>>(agg2, (long long)NN_ * H2_);
  scatter_kernel<H2_><<<(int)(((long long)E_ * H2_) / 256), 256, 0, stream>>>(t2, esrc, edst, ew, agg2);

  // --- Norm2 + leaky -> h2 ---
  graph_norm_leaky_kernel<H2_><<<G_, 256, 0, stream>>>(agg2, inv_dst, gamma2, beta2, alpha2, h2);

  // --- SortPool2 ---
  sort_channels_kernel<H2_><<<NN_, H2_, 0, stream>>>(h2, hs2, key2);
  topk_kernel<<<G_, NPG_, 0, stream>>>(key2, topidx2);
  gather_topk_kernel<H2_><<<(G_ * K_ * H2_) / 256, 256, 0, stream>>>(hs2, topidx2, readout, K_ * H_);

  // --- Classifier (split-K WMMA with atomic reduction) ---
  fill0_kernel<<<32, 256, 0, stream>>>(out, (long long)G_ * OUT_);
  clf_wmma_kernel<<<(16 * 40) / 8, 256, 0, stream>>>(readout, Wc, out);
}